// GCN_60894046322930
// MI455X (gfx1250) — compile-verified
//
#include <hip/hip_runtime.h>

typedef __attribute__((ext_vector_type(2))) float v2f;
typedef __attribute__((ext_vector_type(8))) float v8f;

// LDS-typed scalar: keeps every access on the ds_* path (no flat fallback)
typedef __attribute__((address_space(3))) float lfloat;

#define N_DIN 128
#define N_DH  128
#define N_DOUT 64

// ---------------------------------------------------------------------------
// CDNA5 async global->LDS path (ASYNCcnt-tracked), sync fallback otherwise.
// Signature (from hipcc diagnostic): (v4i AS1*, v4i AS3*, imm i32, imm i32)
// ---------------------------------------------------------------------------
#if defined(__AMDGCN__) && __has_builtin(__builtin_amdgcn_global_load_async_to_lds_b128) && __has_builtin(__builtin_amdgcn_s_wait_asynccnt)
#define USE_ASYNC_LDS 1
#else
#define USE_ASYNC_LDS 0
#endif

typedef int   v4i_t __attribute__((vector_size(16)));
typedef float v4f_t __attribute__((vector_size(16)));
typedef __attribute__((address_space(1))) v4i_t* as1_v4i;
typedef __attribute__((address_space(3))) v4i_t* as3_v4i;
typedef __attribute__((address_space(3))) v4f_t* as3_v4f;

__device__ __forceinline__ void copy16_g2l(const float* __restrict__ g, lfloat* l) {
#if USE_ASYNC_LDS
    __builtin_amdgcn_global_load_async_to_lds_b128((as1_v4i)g, (as3_v4i)l, 0, 0);
#else
    *(as3_v4f)l = *(const v4f_t*)g;
#endif
}

__device__ __forceinline__ void async_wait0() {
#if USE_ASYNC_LDS
    __builtin_amdgcn_s_wait_asynccnt(0);
#endif
}

// ---------------------------------------------------------------------------
// fp32 atomic add, agent scope, relaxed -> global_atomic_add_f32
// ---------------------------------------------------------------------------
__device__ __forceinline__ void atomicAddAgentF32(float* p, float v) {
    __hip_atomic_fetch_add(p, v, __ATOMIC_RELAXED, __HIP_MEMORY_SCOPE_AGENT);
}

// ---------------------------------------------------------------------------
// Degree kernels
// ---------------------------------------------------------------------------
__global__ void deg_init(float* __restrict__ deg, int n) {
    int i = blockIdx.x * blockDim.x + threadIdx.x;
    if (i < n) deg[i] = 1.0f;
}

__global__ void deg_count(const int* __restrict__ dst, float* __restrict__ deg, int E) {
    int e = blockIdx.x * blockDim.x + threadIdx.x;
    if (e < E) atomicAddAgentF32(&deg[dst[e]], 1.0f);
}

__global__ void deg_rsqrt(float* __restrict__ deg, int n) {
    int i = blockIdx.x * blockDim.x + threadIdx.x;
    if (i < n) deg[i] = rsqrtf(deg[i]);
}

// ---------------------------------------------------------------------------
// GEMM: out[M,N] = (RELU? relu(A) : A)[M,K] @ W[K,N] via V_WMMA_F32_16X16X4_F32
//  - workgroup: 64 rows of A, all N cols; wave w owns 16-col tile n0 = 16w
//  - K panelled in 32-wide chunks, double-buffered in LDS via async copies
//  - per panel: batch-preload B (bf[8]) and A (af[4][8]) fragments from LDS,
//    then fire 32 WMMAs back-to-back, round-robin over 4 accumulators
// ---------------------------------------------------------------------------
template<int RELU>
__global__ void gemm_wmma_f32(const float* __restrict__ A, const float* __restrict__ W,
                              float* __restrict__ out, int M, int K, int N) {
    constexpr int KP  = 32;
    constexpr int PAD = 4;                       // stride 36 floats: 16B-aligned
    extern __shared__ float lds_raw[];
    lfloat* lds = (lfloat*)lds_raw;
    const int BUF = KP * N + 64 * (KP + PAD);

    const int tid  = threadIdx.x;
    const int nthr = blockDim.x;
    const int m0   = blockIdx.x * 64;
    const int lane = tid & 31;
    const int wave = tid >> 5;
    const int half = lane >> 4;
    const int lr   = lane & 15;
    const int n0   = wave * 16;

    lfloat* sW0 = lds;              lfloat* sA0 = lds + KP * N;
    lfloat* sW1 = lds + BUF;        lfloat* sA1 = lds + BUF + KP * N;

    auto stage = [&](int kp, lfloat* sW, lfloat* sA) {
        // W panel: KP*N contiguous floats starting at row kp
        const float* wsrc = W + (size_t)kp * N;
        const int nW = (KP * N) >> 2;                    // 16B chunks
        for (int i = tid; i < nW; i += nthr)
            copy16_g2l(wsrc + (i << 2), sW + (i << 2));
        // A panel: 64 rows x KP floats (row clamped at M-1; stores guarded)
        const int cpr = KP >> 2;                         // chunks per row
        const int nA = 64 * cpr;
        for (int i = tid; i < nA; i += nthr) {
            int r  = i / cpr;
            int c4 = (i - r * cpr) << 2;
            int gr = m0 + r; if (gr >= M) gr = M - 1;
            copy16_g2l(A + (size_t)gr * K + kp + c4, sA + r * (KP + PAD) + c4);
        }
    };

    v8f acc0 = {}, acc1 = {}, acc2 = {}, acc3 = {};
    const int nPanels = K / KP;

    stage(0, sW0, sA0);
    async_wait0();
    __syncthreads();

    for (int p = 0; p < nPanels; ++p) {
        lfloat* cW = (p & 1) ? sW1 : sW0;
        lfloat* cA = (p & 1) ? sA1 : sA0;
        if (p + 1 < nPanels)
            stage((p + 1) * KP, ((p + 1) & 1) ? sW1 : sW0, ((p + 1) & 1) ? sA1 : sA0);

        // --- batch preload B fragments (shared across all 4 row-subtiles) ---
        // B 4x16 f32 layout: VGPR0 = rows {k | k+2}, VGPR1 = rows {k+1 | k+3}
        v2f bf[8];
        #pragma unroll
        for (int kk = 0; kk < 8; ++kk) {
            bf[kk].x = cW[(kk * 4 + 2 * half) * N + n0 + lr];
            bf[kk].y = cW[(kk * 4 + 2 * half + 1) * N + n0 + lr];
        }
        // --- batch preload A fragments for all four 16-row subtiles ---
        // A 16x4 f32 layout: VGPR0 = K {k | k+2}, VGPR1 = K {k+1 | k+3}
        v2f af[4][8];
        #pragma unroll
        for (int ms = 0; ms < 4; ++ms) {
            lfloat* aRow = cA + (ms * 16 + lr) * (KP + PAD);
            #pragma unroll
            for (int kk = 0; kk < 8; ++kk) {
                v2f a;
                a.x = aRow[kk * 4 + 2 * half];
                a.y = aRow[kk * 4 + 2 * half + 1];
                if (RELU) { a.x = fmaxf(a.x, 0.0f); a.y = fmaxf(a.y, 0.0f); }
                af[ms][kk] = a;
            }
        }
        // --- 32 back-to-back WMMAs, round-robin over independent accumulators ---
        #pragma unroll
        for (int kk = 0; kk < 8; ++kk) {
            acc0 = __builtin_amdgcn_wmma_f32_16x16x4_f32(
                    false, af[0][kk], false, bf[kk], (short)0, acc0, false, false);
            acc1 = __builtin_amdgcn_wmma_f32_16x16x4_f32(
                    false, af[1][kk], false, bf[kk], (short)0, acc1, false, false);
            acc2 = __builtin_amdgcn_wmma_f32_16x16x4_f32(
                    false, af[2][kk], false, bf[kk], (short)0, acc2, false, false);
            acc3 = __builtin_amdgcn_wmma_f32_16x16x4_f32(
                    false, af[3][kk], false, bf[kk], (short)0, acc3, false, false);
        }
        if (p + 1 < nPanels) { async_wait0(); __syncthreads(); }
    }

    // C/D layout: VGPR v -> row (v + 8*half), col = n0 + lr
    #pragma unroll
    for (int ms = 0; ms < 4; ++ms) {
        v8f acc = (ms == 0) ? acc0 : (ms == 1) ? acc1 : (ms == 2) ? acc2 : acc3;
        #pragma unroll
        for (int v = 0; v < 8; ++v) {
            int gr = m0 + ms * 16 + v + 8 * half;
            if (gr < M) out[(size_t)gr * N + n0 + lr] = acc[v];
        }
    }
}

// ---------------------------------------------------------------------------
// node_init: out[i][c] = bias[c] + h[i][c] * dinv[i]^2  (self-loop + bias,
// fully initializes the poisoned output before edge atomics)
// ---------------------------------------------------------------------------
__global__ void node_init(const float* __restrict__ h, const float* __restrict__ dinv,
                          const float* __restrict__ bias, float* __restrict__ out,
                          int n, int C) {
    int idx = blockIdx.x * blockDim.x + threadIdx.x;
    int perNode = C >> 2;
    int total = n * perNode;
    if (idx >= total) return;
    int node = idx / perNode;
    int c4 = (idx - node * perNode) * 4;
    float di = dinv[node];
    float s = di * di;
    float4 hv = *(const float4*)(h + (size_t)node * C + c4);
    float4 o;
    o.x = bias[c4 + 0] + hv.x * s;
    o.y = bias[c4 + 1] + hv.y * s;
    o.z = bias[c4 + 2] + hv.z * s;
    o.w = bias[c4 + 3] + hv.w * s;
    *(float4*)(out + (size_t)node * C + c4) = o;
}

// ---------------------------------------------------------------------------
// edge_agg: warp per edge; float4 gather (C==128 fast path), fp32 atomic
// scatter-add into out[dst] (destination fits in 192MB L2)
// ---------------------------------------------------------------------------
__global__ void edge_agg(const float* __restrict__ h, const int* __restrict__ src,
                         const int* __restrict__ dst, const float* __restrict__ dinv,
                         float* __restrict__ out, int E, int C) {
    int gid = blockIdx.x * blockDim.x + threadIdx.x;
    int e = gid >> 5;
    int lane = gid & 31;
    if (e >= E) return;
    int s = src[e], d = dst[e];
    float nrm = dinv[s] * dinv[d];
    const float* hs = h + (size_t)s * C;
    float* od = out + (size_t)d * C;
    if (C == 128) {
        float4 hv = *(const float4*)(hs + lane * 4);
        float* p = od + lane * 4;
        atomicAddAgentF32(p + 0, hv.x * nrm);
        atomicAddAgentF32(p + 1, hv.y * nrm);
        atomicAddAgentF32(p + 2, hv.z * nrm);
        atomicAddAgentF32(p + 3, hv.w * nrm);
    } else {
        for (int c = lane; c < C; c += 32)
            atomicAddAgentF32(&od[c], hs[c] * nrm);
    }
}

// ---------------------------------------------------------------------------
extern "C" void kernel_launch(void* const* d_in, const int* in_sizes, int n_in,
                              void* d_out, int out_size, void* d_ws, size_t ws_size,
                              hipStream_t stream) {
    const float* x  = (const float*)d_in[0];
    const int*   ei = (const int*)  d_in[1];
    const float* W1 = (const float*)d_in[2];
    const float* b1 = (const float*)d_in[3];
    const float* W2 = (const float*)d_in[4];
    const float* b2 = (const float*)d_in[5];
    const float* W3 = (const float*)d_in[6];
    const float* b3 = (const float*)d_in[7];
    float* out = (float*)d_out;

    const int n = in_sizes[0] / N_DIN;   // 100000
    const int E = in_sizes[1] / 2;       // 1600000
    const int* src = ei;
    const int* dst = ei + E;

    // workspace layout: dinv | bufA | bufB
    char* wsb = (char*)d_ws;
    float* dinv = (float*)wsb;
    size_t off = ((size_t)n * sizeof(float) + 511) & ~(size_t)511;
    size_t bufBytes = (((size_t)n * N_DH * sizeof(float)) + 511) & ~(size_t)511;
    float* bufA = (float*)(wsb + off);
    float* bufB = (float*)(wsb + off + bufBytes);

    // ---- degrees -> dinv ----
    deg_init <<<(n + 255) / 256, 256, 0, stream>>>(dinv, n);
    deg_count<<<(E + 255) / 256, 256, 0, stream>>>(dst, dinv, E);
    deg_rsqrt<<<(n + 255) / 256, 256, 0, stream>>>(dinv, n);

    const int mBlocks = (n + 63) / 64;
    const long long edgeThreads = (long long)E * 32;
    const int edgeBlocks = (int)((edgeThreads + 255) / 256);

    auto smemFor = [](int N) -> size_t {
        return (size_t)2 * (32 * N + 64 * 36) * sizeof(float);  // double buffer
    };

    // ---- layer 1: H1 = X @ W1; agg -> bufB ----
    {
        const int K = N_DIN, N = N_DH;
        gemm_wmma_f32<0><<<mBlocks, 32 * (N / 16), smemFor(N), stream>>>(x, W1, bufA, n, K, N);
        node_init<<<((n * (N / 4)) + 255) / 256, 256, 0, stream>>>(bufA, dinv, b1, bufB, n, N);
        edge_agg<<<edgeBlocks, 256, 0, stream>>>(bufA, src, dst, dinv, bufB, E, N);
    }
    // ---- layer 2: H2 = relu(bufB) @ W2; agg -> bufB ----
    {
        const int K = N_DH, N = N_DH;
        gemm_wmma_f32<1><<<mBlocks, 32 * (N / 16), smemFor(N), stream>>>(bufB, W2, bufA, n, K, N);
        node_init<<<((n * (N / 4)) + 255) / 256, 256, 0, stream>>>(bufA, dinv, b2, bufB, n, N);
        edge_agg<<<edgeBlocks, 256, 0, stream>>>(bufA, src, dst, dinv, bufB, E, N);
    }
    // ---- layer 3: H3 = relu(bufB) @ W3; agg -> d_out ----
    {
        const int K = N_DH, N = N_DOUT;
        gemm_wmma_f32<1><<<mBlocks, 32 * (N / 16), smemFor(N), stream>>>(bufB, W3, bufA, n, K, N);
        node_init<<<((n * (N / 4)) + 255) / 256, 256, 0, stream>>>(bufA, dinv, b3, out, n, N);
        edge_agg<<<edgeBlocks, 256, 0, stream>>>(bufA, src, dst, dinv, out, E, N);
    }
}